// SparseHashedNNDistance_39857296507461
// MI455X (gfx1250) — compile-verified
//
#include <hip/hip_runtime.h>
#include <math.h>

// Problem constants (from reference)
#define NB     2
#define NPTS   5000
#define NF     64
#define BINSZ  500
#define NBINS  10
#define HALF   5        // nbins // 2 rotation columns used
#define ROTCOLS 100     // MAX_NUM_BINS // 2 (row stride of rotations)
#define KNN    5
#define PADROWS 512     // 500 padded to multiple of 16
#define LSTR   66       // LDS row stride in floats (bank-conflict padding, keeps 8B align)

typedef __attribute__((ext_vector_type(2))) float v2f;
typedef __attribute__((ext_vector_type(4))) float v4f;
typedef __attribute__((ext_vector_type(8))) float v8f;

// ---------------------------------------------------------------------------
// Kernel 1: zero the 200MB output (dominant bandwidth cost). Nontemporal
// 128-bit streaming stores — output (200MB) exceeds the 192MB L2.
// ---------------------------------------------------------------------------
__global__ void zero_out_kernel(float* __restrict__ out, long n) {
  long i = (long)blockIdx.x * blockDim.x + threadIdx.x;
  long stride = (long)gridDim.x * blockDim.x;
  long n4 = n >> 2;
  v4f* o4 = (v4f*)out;
  v4f z = {0.f, 0.f, 0.f, 0.f};
  for (long j = i; j < n4; j += stride)
    __builtin_nontemporal_store(z, &o4[j]);
  for (long j = (n4 << 2) + i; j < n; j += stride)
    out[j] = 0.f;
}

// ---------------------------------------------------------------------------
// Kernel 2: LSH bin assignment. One thread per point: 5 dot products over 64
// features, argmax over [m, -m] with first-max (jnp.argmax) semantics.
// ---------------------------------------------------------------------------
__global__ void assign_bins_kernel(const float* __restrict__ pts,
                                   const float* __restrict__ rot,
                                   int* __restrict__ bin_idx) {
  int t = blockIdx.x * blockDim.x + threadIdx.x;
  if (t >= NB * NPTS) return;
  const float* p = pts + (size_t)t * NF;
  float m[HALF] = {0.f, 0.f, 0.f, 0.f, 0.f};
  for (int f = 0; f < NF; ++f) {
    float pv = p[f];
#pragma unroll
    for (int h = 0; h < HALF; ++h) m[h] += pv * rot[f * ROTCOLS + h];
  }
  float best = m[0];
  int bi = 0;
#pragma unroll
  for (int h = 1; h < HALF; ++h)
    if (m[h] > best) { best = m[h]; bi = h; }
#pragma unroll
  for (int h = 0; h < HALF; ++h)
    if (-m[h] > best) { best = -m[h]; bi = HALF + h; }
  bin_idx[t] = bi;
}

// ---------------------------------------------------------------------------
// Kernel 3: stable counting sort by bin (== stable argsort of bin_idx).
// One wave32 block per batch; thread t owns bin t.
// ---------------------------------------------------------------------------
__global__ void bin_sort_kernel(const int* __restrict__ bin_idx,
                                int* __restrict__ perm) {
  __shared__ int cnt[NBINS], off[NBINS];
  int b = blockIdx.x;
  const int* bi = bin_idx + b * NPTS;
  int t = threadIdx.x;
  if (t < NBINS) {
    int c = 0;
    for (int i = 0; i < NPTS; ++i) c += (bi[i] == t);
    cnt[t] = c;
  }
  __syncthreads();
  if (t == 0) {
    int s = 0;
    for (int h = 0; h < NBINS; ++h) { off[h] = s; s += cnt[h]; }
  }
  __syncthreads();
  if (t < NBINS) {
    int o = off[t];
    for (int i = 0; i < NPTS; ++i)
      if (bi[i] == t) perm[b * NPTS + (o++)] = i;
  }
}

// ---------------------------------------------------------------------------
// Kernel 4: per (batch, bin) Gram matrix via V_WMMA_F32_16X16X4_F32,
// fused score transform + running top-5, lane merge through LDS, final
// exp(-0.1*sqrt(d2)) only on winners, direct stores to the zeroed output.
//
// LDS layout (dynamic, 221184 B total, under the 320KB WGP pool):
//   Plds  : PADROWS*LSTR floats   (padded point tile, row-major)
//   nrm   : PADROWS floats        (row squared norms)
//   mval  : 8*16*16*KNN floats    (per-wave top-k merge scratch, values)
//   midx  : 8*16*16*KNN ints      (per-wave top-k merge scratch, indices)
//   lperm : PADROWS ints          (local->global point ids)
// ---------------------------------------------------------------------------
__global__ void __launch_bounds__(256) bin_gram_topk_kernel(
    const float* __restrict__ pts, const int* __restrict__ perm,
    float* __restrict__ out) {
  extern __shared__ float smem[];
  float* Plds  = smem;                              // 33792 floats
  float* nrm   = Plds + PADROWS * LSTR;             // 512
  float* mval  = nrm + PADROWS;                     // 10240
  int*   midx  = (int*)(mval + 8 * 16 * 16 * KNN);  // 10240
  int*   lperm = (int*)(midx + 8 * 16 * 16 * KNN);  // 512

  const int b = blockIdx.x / NBINS;
  const int g = blockIdx.x % NBINS;
  const int tid = threadIdx.x;
  const int* pm = perm + b * NPTS + g * BINSZ;

  for (int r = tid; r < PADROWS; r += 256) lperm[r] = (r < BINSZ) ? pm[r] : 0;
  __syncthreads();

  // Gather bin's points into padded LDS tile (rows >= 500 are zero, excluded
  // from top-k via column masking).
  for (int i = tid; i < PADROWS * NF; i += 256) {
    int r = i >> 6, k = i & 63;
    float v = 0.f;
    if (r < BINSZ) v = pts[((size_t)b * NPTS + lperm[r]) * NF + k];
    Plds[r * LSTR + k] = v;
  }
  __syncthreads();

  // Row squared norms.
  for (int r = tid; r < PADROWS; r += 256) {
    const float* row = &Plds[r * LSTR];
    float s = 0.f;
    for (int k = 0; k < NF; ++k) s += row[k] * row[k];
    nrm[r] = s;
  }
  __syncthreads();

  const int w = tid >> 5, lane = tid & 31;
  const int lo = lane & 15, hi = lane >> 4;

  // 32 row-strips of 16; 8 waves -> 4 strips each (uniform loop so the
  // trailing __syncthreads is legal and EXEC is all-1s at every WMMA).
  for (int iter = 0; iter < 4; ++iter) {
    const int rowBase = (iter * 8 + w) * 16;

    // Preload A strip per ISA 32-bit 16x4 A layout:
    // lane(lo,hi) holds A[M=lo][K = 4*kk + 2*hi + {0,1}] in a[kk].{x,y}
    v2f a[16];
    {
      const float* ap = &Plds[(rowBase + lo) * LSTR + 2 * hi];
#pragma unroll
      for (int kk = 0; kk < 16; ++kk) { a[kk].x = ap[4 * kk]; a[kk].y = ap[4 * kk + 1]; }
    }

    // Running per-lane top-5 for the 8 rows this lane covers in C.
    float tval[8][KNN];
    int   tidx[8][KNN];
#pragma unroll
    for (int q = 0; q < 8; ++q)
#pragma unroll
      for (int e = 0; e < KNN; ++e) { tval[q][e] = -3.0e38f; tidx[q][e] = 0; }

    for (int ct = 0; ct < 32; ++ct) {
      const int colBase = ct * 16;
      v8f c = {0.f, 0.f, 0.f, 0.f, 0.f, 0.f, 0.f, 0.f};
      const float* bp = &Plds[(colBase + lo) * LSTR + 2 * hi];
#pragma unroll
      for (int kk = 0; kk < 16; ++kk) {
        v2f bb;
        bb.x = bp[4 * kk];
        bb.y = bp[4 * kk + 1];
        // D = A(16x4) * B(4x16) + C, full f32
        c = __builtin_amdgcn_wmma_f32_16x16x4_f32(false, a[kk], false, bb,
                                                  (short)0, c, false, false);
      }
      const int col = colBase + lo;             // C layout: N = lane&15
      const bool valid = (col < BINSZ);
      const float nj = nrm[col];
#pragma unroll
      for (int q = 0; q < 8; ++q) {             // C layout: M = q + 8*hi
        // score s = 2*gram - |p_j|^2 ; maximizing s == maximizing affinity
        float v = valid ? (2.f * c[q] - nj) : -3.0e38f;
        int jj = col;
#pragma unroll
        for (int e = 0; e < KNN; ++e) {         // branchless sorted insert
          bool sw = v > tval[q][e];
          float ov = tval[q][e]; int oi = tidx[q][e];
          tval[q][e] = sw ? v : ov; tidx[q][e] = sw ? jj : oi;
          v = sw ? ov : v; jj = sw ? oi : jj;
        }
      }
    }

    // Dump per-lane lists to this wave's private LDS scratch region.
    float* wv = mval + w * 16 * 16 * KNN;
    int*   wi = midx + w * 16 * 16 * KNN;
#pragma unroll
    for (int q = 0; q < 8; ++q) {
      int rr = q + 8 * hi;                      // row within strip
      int base = (rr * 16 + lo) * KNN;
#pragma unroll
      for (int e = 0; e < KNN; ++e) { wv[base + e] = tval[q][e]; wi[base + e] = tidx[q][e]; }
    }
    // CDNA5 split counters: drain LDS ops, then a wave-level ordering fence.
    asm volatile("s_wait_dscnt 0" ::: "memory");
    __builtin_amdgcn_wave_barrier();

    // Lanes 0..15: merge 80 candidates -> final top-5 for one row, emit.
    if (lane < 16) {
      const int rloc = rowBase + lo;
      if (rloc < BINSZ) {
        float bv[KNN]; int bj[KNN];
#pragma unroll
        for (int e = 0; e < KNN; ++e) { bv[e] = -3.4e38f; bj[e] = 0x7fffffff; }
        const float* rv = wv + lo * 16 * KNN;
        const int*   rj = wi + lo * 16 * KNN;
        for (int e = 0; e < 16 * KNN; ++e) {
          float v = rv[e]; int jj = rj[e];
#pragma unroll
          for (int p = 0; p < KNN; ++p) {       // ties -> smaller index (top_k)
            bool sw = (v > bv[p]) || (v == bv[p] && jj < bj[p]);
            float ov = bv[p]; int oi = bj[p];
            bv[p] = sw ? v : ov; bj[p] = sw ? jj : oi;
            v = sw ? ov : v; jj = sw ? oi : jj;
          }
        }
        const float ni = nrm[rloc];
        const int src = lperm[rloc];
        float* orow = out + (size_t)b * NPTS * NPTS + (size_t)src * NPTS;
#pragma unroll
        for (int e = 0; e < KNN; ++e) {
          float d2 = fmaxf(ni - bv[e], 1e-6f);  // d2 = |pi|^2 + |pj|^2 - 2g
          float aff = expf(-0.1f * sqrtf(d2));
          orow[lperm[bj[e]]] = aff;             // unique (src,dst): plain store
        }
      }
    }
    __syncthreads();
  }
}

// ---------------------------------------------------------------------------
// Host launcher
// ---------------------------------------------------------------------------
extern "C" void kernel_launch(void* const* d_in, const int* in_sizes, int n_in,
                              void* d_out, int out_size, void* d_ws, size_t ws_size,
                              hipStream_t stream) {
  const float* pts = (const float*)d_in[0];   // [2, 5000, 64] f32
  const float* rot = (const float*)d_in[1];   // [64, 100] f32
  float* out = (float*)d_out;                 // [2, 5000, 5000] f32

  int* bin_idx = (int*)d_ws;                  // [2*5000]
  int* perm    = bin_idx + NB * NPTS;         // [2*5000]

  zero_out_kernel<<<4096, 256, 0, stream>>>(out, (long)out_size);
  assign_bins_kernel<<<(NB * NPTS + 255) / 256, 256, 0, stream>>>(pts, rot, bin_idx);
  bin_sort_kernel<<<NB, 32, 0, stream>>>(bin_idx, perm);

  size_t lds_bytes =
      (size_t)(PADROWS * LSTR + PADROWS + 8 * 16 * 16 * KNN) * sizeof(float) +
      (size_t)(8 * 16 * 16 * KNN + PADROWS) * sizeof(int);   // 221184 B
  bin_gram_topk_kernel<<<NB * NBINS, 256, lds_bytes, stream>>>(pts, perm, out);
}